// PureGCN_v1_56521769616157
// MI455X (gfx1250) — compile-verified
//
#include <hip/hip_runtime.h>

// Problem constants (match reference)
#define DIN 256
#define H   128
#define LN_EPS 1e-5f

typedef float v2f  __attribute__((ext_vector_type(2)));
typedef float v8f  __attribute__((ext_vector_type(8)));
typedef unsigned int u32x4 __attribute__((ext_vector_type(4)));
typedef int   i32x8 __attribute__((ext_vector_type(8)));
typedef int   i32x4 __attribute__((ext_vector_type(4)));

// ---------------------------------------------------------------------------
// Utility kernels
// ---------------------------------------------------------------------------
__global__ void zero_f4_kernel(float4* __restrict__ p, long n4) {
    long i = (long)blockIdx.x * blockDim.x + threadIdx.x;
    if (i < n4) p[i] = make_float4(0.f, 0.f, 0.f, 0.f);
}

__global__ void deg_count_kernel(const int* __restrict__ dst, float* __restrict__ deg, int E) {
    int e = blockIdx.x * blockDim.x + threadIdx.x;
    if (e < E) atomicAdd(&deg[dst[e]], 1.0f);
}

__global__ void norm_finalize_kernel(float* __restrict__ norm, int N) {
    int i = blockIdx.x * blockDim.x + threadIdx.x;
    if (i < N) norm[i] = rsqrtf(1.0f + norm[i]);
}

// ---------------------------------------------------------------------------
// GEMM: h0 = x @ W + b  via V_WMMA_F32_16X16X4_F32 (full fp32 precision).
// The 16x256 x-tile is DMA'd to LDS by the Tensor Data Mover (wave 0 issues
// TENSOR_LOAD_TO_LDS, waits on TENSORcnt, block barrier releases the rest).
// Fused epilogue writes h0 (residual), t = norm*h0 (scatter gather source)
// and seeds the scatter accumulator with t (so "agg + h" comes for free).
// N must be a multiple of 16 (100000 = 6250 * 16).
// ---------------------------------------------------------------------------
__global__ __launch_bounds__(256) void gemm_wmma_f32_kernel(
        const float* __restrict__ x, const float* __restrict__ W,
        const float* __restrict__ b, const float* __restrict__ norm,
        float* __restrict__ ori, float* __restrict__ cur,
        float* __restrict__ agg, int N) {
    __shared__ float As[16 * DIN];          // 16 rows x 256 cols = 16 KB, LDS offset 0
    const int rowBase = blockIdx.x * 16;
    const int tid = threadIdx.x;

    // --- TDM: 2D tile load, 16 rows x 256 f32, row stride 256 elements ---
    if (tid < 32) {                          // wave 0 issues the DMA (EXEC ignored)
        const unsigned long long ga = (unsigned long long)(x + (size_t)rowBase * DIN);
        u32x4 g0;
        g0[0] = 1u;                          // count=1, user descriptor, no gather
        g0[1] = 0u;                          // lds_addr = 0 (As is the only LDS alloc)
        g0[2] = (unsigned)(ga & 0xffffffffu);        // global_addr[31:0] (tile start)
        g0[3] = (unsigned)((ga >> 32) & 0x1ffffffu)  // global_addr[56:32]
                | (2u << 30);                        // type = 2 ("image")
        i32x8 g1;
        g1[0] = (2 << 16);                   // workgroup_mask=0, data_size=2 (4B)
        g1[1] = (DIN & 0xffff) << 16;        // tensor_dim0[15:0]=256 in bits[31:16]
        g1[2] = (16 & 0xffff) << 16;         // tensor_dim0[31:16]=0 | tensor_dim1 lo=16
        g1[3] = (DIN & 0xffff) << 16;        // tensor_dim1 hi=0 | tile_dim0=256
        g1[4] = 16;                          // tile_dim1=16, tile_dim2=0
        g1[5] = DIN;                         // tensor_dim0_stride[31:0]=256
        g1[6] = 0;                           // stride0 hi | stride1 lo (2D: unused)
        g1[7] = 0;
        i32x4 gz4 = {0, 0, 0, 0};            // groups 2/3 unused for 2D tensors
        i32x8 gz8 = {0, 0, 0, 0, 0, 0, 0, 0};
        // 6-arg form (amdgpu-toolchain clang-23): (g0, g1, g2, g3, g4, cpol)
        __builtin_amdgcn_tensor_load_to_lds(g0, g1, gz4, gz4, gz8, 0);
        __builtin_amdgcn_s_wait_tensorcnt(0);
    }
    __syncthreads();

    const int wave    = tid >> 5;           // 0..7 -> 16-col tile
    const int lane    = tid & 31;
    const int colBase = wave * 16;
    const int lhalf   = lane >> 4;          // 0: lanes 0-15, 1: lanes 16-31
    const int l15     = lane & 15;
    const int koff    = lhalf * 2;          // upper half holds K+2, K+3

    const float* As_row = As + l15 * DIN;   // A row M = l15 (both halves)
    v8f c = {};

    #pragma unroll 4
    for (int k = 0; k < DIN; k += 4) {
        v2f a, bm;
        // A 16x4 fp32 fragment: VGPR0 = K=k(+2), VGPR1 = K=k+1(+3)
        a.x = As_row[k + koff];
        a.y = As_row[k + koff + 1];
        // B 4x16 fp32 fragment: row K across lanes (mirrored K split)
        bm.x = W[(size_t)(k + koff) * H + colBase + l15];
        bm.y = W[(size_t)(k + koff + 1) * H + colBase + l15];
        // 8 args: (neg_a, A, neg_b, B, c_mod, C, reuse_a, reuse_b)
        c = __builtin_amdgcn_wmma_f32_16x16x4_f32(false, a, false, bm,
                                                  (short)0, c, false, false);
    }

    const float bias = b[colBase + l15];
    #pragma unroll
    for (int r = 0; r < 8; ++r) {
        const int m = r + lhalf * 8;        // C/D layout: VGPR r -> row r (+8)
        const size_t idx = (size_t)(rowBase + m) * H + colBase + l15;
        const float h0 = c[r] + bias;
        const float t  = norm[rowBase + m] * h0;
        ori[idx] = h0;                       // residual source
        cur[idx] = t;                        // scatter gather source
        agg[idx] = t;                        // accumulator seed (= "+ h" term)
    }
}

// ---------------------------------------------------------------------------
// Scatter: agg[dst] += cur[src]  (segment_sum). One wave per edge,
// each lane moves 4 consecutive features (float4 gather, 4 f32 atomics).
// ---------------------------------------------------------------------------
__global__ void scatter_kernel(const float* __restrict__ t, float* __restrict__ agg,
                               const int* __restrict__ esrc, const int* __restrict__ edst,
                               int E) {
    long gid = (long)blockIdx.x * blockDim.x + threadIdx.x;
    long e = gid >> 5;
    int lane = (int)(gid & 31);
    if (e >= E) return;
    int s = esrc[e], d = edst[e];
    float4 v = ((const float4*)(t + (size_t)s * H))[lane];
    float* ap = agg + (size_t)d * H + lane * 4;
    atomicAdd(ap + 0, v.x);
    atomicAdd(ap + 1, v.y);
    atomicAdd(ap + 2, v.z);
    atomicAdd(ap + 3, v.w);
}

// ---------------------------------------------------------------------------
// Fused layer-0 finish + layer-1 entry:
//   h   = norm*agg                (layer-0 output)
//   v   = h + ori                 (residual)
//   y   = relu(LayerNorm(v))      (gamma/beta of layer 1)
//   t1  = norm*y  -> written to BOTH cur (gather source) and agg (re-seed)
// One 128-thread block per row (4 wave32, LDS tree reduction).
// ---------------------------------------------------------------------------
__global__ __launch_bounds__(H) void ln_relu_scale_kernel(
        const float* __restrict__ aggr, const float* __restrict__ ori,
        const float* __restrict__ gamma, const float* __restrict__ beta,
        const float* __restrict__ norm, float* __restrict__ cur,
        float* __restrict__ aggw) {
    const int row = blockIdx.x;
    const int f = threadIdx.x;
    __shared__ float s1[H], s2[H];

    const float nv = norm[row];
    const size_t idx = (size_t)row * H + f;
    const float v = nv * aggr[idx] + ori[idx];
    s1[f] = v;
    s2[f] = v * v;
    __syncthreads();
    #pragma unroll
    for (int off = H / 2; off > 0; off >>= 1) {
        if (f < off) { s1[f] += s1[f + off]; s2[f] += s2[f + off]; }
        __syncthreads();
    }
    const float mu  = s1[0] * (1.0f / H);
    const float var = s2[0] * (1.0f / H) - mu * mu;   // E[x^2] - mu^2
    float y = gamma[f] * (v - mu) * rsqrtf(var + LN_EPS) + beta[f];
    y = fmaxf(y, 0.0f);
    const float t = nv * y;
    cur[idx]  = t;        // gather source for layer-1 scatter
    aggw[idx] = t;        // accumulator seed (aggw aliases aggr; per-thread element)
}

// ---------------------------------------------------------------------------
// Final: out *= norm[row]   (in place on the accumulator = d_out)
// ---------------------------------------------------------------------------
__global__ void final_scale_kernel(float4* __restrict__ out, const float* __restrict__ norm,
                                   long n4) {
    long i = (long)blockIdx.x * blockDim.x + threadIdx.x;
    if (i >= n4) return;
    float nv = norm[i >> 5];                // row = i / (H/4)
    float4 v = out[i];
    out[i] = make_float4(nv * v.x, nv * v.y, nv * v.z, nv * v.w);
}

// ---------------------------------------------------------------------------
// Host-side orchestration
// ---------------------------------------------------------------------------
extern "C" void kernel_launch(void* const* d_in, const int* in_sizes, int n_in,
                              void* d_out, int out_size, void* d_ws, size_t ws_size,
                              hipStream_t stream) {
    const float* x     = (const float*)d_in[0];   // [N, 256]
    const float* W     = (const float*)d_in[1];   // [256, 128]
    const float* bvec  = (const float*)d_in[2];   // [128]
    const float* gamma = (const float*)d_in[3];   // [2, 128]
    const float* beta  = (const float*)d_in[4];   // [2, 128]
    const int*   esrc  = (const int*)d_in[5];     // [E]
    const int*   edst  = (const int*)d_in[6];     // [E]

    const int N = in_sizes[0] / DIN;
    const int E = in_sizes[5];
    const long NH  = (long)N * H;
    const long NH4 = NH / 4;

    // workspace layout: norm [N] | ori [N*H] | cur [N*H]
    float* norm = (float*)d_ws;
    const long normPad = ((long)N + 63) & ~63L;
    float* ori = norm + normPad;
    float* cur = ori + NH;
    float* agg = (float*)d_out;                   // d_out doubles as accumulator

    const int TPB = 256;
    const long ewGrid = ((long)E * 32 + TPB - 1) / TPB;   // wave-per-edge scatter
    const long e4Grid = (NH4 + TPB - 1) / TPB;            // elementwise float4 grids

    // degree -> norm = rsqrt(1 + deg)
    zero_f4_kernel<<<(normPad / 4 + TPB - 1) / TPB, TPB, 0, stream>>>((float4*)norm, normPad / 4);
    deg_count_kernel<<<(E + TPB - 1) / TPB, TPB, 0, stream>>>(edst, norm, E);
    norm_finalize_kernel<<<(N + TPB - 1) / TPB, TPB, 0, stream>>>(norm, N);

    // h0 = x @ W + b (fp32 WMMA, TDM tile load); also writes t and seeds agg
    gemm_wmma_f32_kernel<<<N / 16, 256, 0, stream>>>(x, W, bvec, norm, ori, cur, agg, N);

    // ---- layer 0 scatter (accumulates onto the seeded agg) ----
    scatter_kernel<<<ewGrid, TPB, 0, stream>>>(cur, agg, esrc, edst, E);

    // ---- fused layer-0 finish + layer-1 residual/LN/ReLU/scale + re-seed ----
    ln_relu_scale_kernel<<<N, H, 0, stream>>>(agg, ori, gamma + H, beta + H, norm, cur, agg);

    // ---- layer 1 scatter ----
    scatter_kernel<<<ewGrid, TPB, 0, stream>>>(cur, agg, esrc, edst, E);

    // ---- final: out = norm * agg (in place on d_out) ----
    final_scale_kernel<<<e4Grid, TPB, 0, stream>>>((float4*)d_out, norm, NH4);
}